// Attention_1838246003374
// MI455X (gfx1250) — compile-verified
//
#include <hip/hip_runtime.h>
#include <math.h>

// ---------------------------------------------------------------------------
// MI455X (gfx1250) attention: fp32 path using V_WMMA_F32_16X16X4_F32,
// async-to-LDS staging (ASYNCcnt) for the GEMM B operand, global prefetch
// for the A stream.
// ---------------------------------------------------------------------------

typedef float v2f __attribute__((ext_vector_type(2)));
typedef float v8f __attribute__((ext_vector_type(8)));

#define HID   4096   // NUM_HEADS * HEAD_DIM
#define KVD   1024   // NUM_KV_HEADS * HEAD_DIM
#define NH    32
#define NKV   8
#define HD    128
#define SEQ   2048
#define BATCH 2

__device__ __forceinline__ v8f wmma_f32(v2f a, v2f b, v8f c) {
  // 8 args: (neg_a, A, neg_b, B, c_mod, C, reuse_a, reuse_b)
  return __builtin_amdgcn_wmma_f32_16x16x4_f32(
      false, a, false, b, (short)0, c, false, false);
}

// Async copy 16B global -> LDS (gfx1250 GLOBAL_LOAD_ASYNC_TO_LDS_B128,
// tracked by ASYNCcnt). The generic pointer to a __shared__ object carries
// the wave-relative LDS offset in its low 32 bits (LDS aperture layout).
__device__ __forceinline__ void async_b128_to_lds(void* lds_dst, const void* gsrc) {
  const unsigned int l = (unsigned int)(unsigned long long)lds_dst;
  asm volatile("global_load_async_to_lds_b128 %0, %1, off"
               :: "v"(l), "v"(gsrc)
               : "memory");
}

__device__ __forceinline__ void wait_async_le1() {
  asm volatile("s_wait_asynccnt 0x1" ::: "memory");
}
__device__ __forceinline__ void wait_async_le0() {
  asm volatile("s_wait_asynccnt 0x0" ::: "memory");
}

// ---------------------------------------------------------------------------
// C[M,N] = A[M,K] @ B[K,N], all row-major fp32.
// Block = 256 threads (8 waves). Block tile = 128(M) x 64(N); wave w owns
// rows [row0 + 16w, +16) and all 64 columns, so the B tile staged in LDS is
// reused by all 8 waves. B is double-buffered: a 16(K) x 64(N) chunk (4 KB)
// is brought in with one async b128 per thread while the previous chunk is
// consumed from LDS.
//
// WMMA fragment layout (V_WMMA_F32_16X16X4_F32):
//   A 16x4: lane m = l&15, k = 2*(l>>4) + {0,1}  (contiguous float2 from A)
//   B 4x16: lane n = l&15, k = 2*(l>>4) + {0,1}  (two LDS reads, stride 256B)
//   D 16x16: VGPR j holds row m = j + 8*(l>>4), col n = l&15
// ---------------------------------------------------------------------------
__global__ void gemm_f32_wmma(const float* __restrict__ A,
                              const float* __restrict__ B,
                              float* __restrict__ C,
                              int M, int N, int K) {
  const int lane = threadIdx.x & 31;
  const int wave = threadIdx.x >> 5;
  const int n16  = lane & 15;
  const int hi   = lane >> 4;
  const int row0 = blockIdx.y * 128 + wave * 16;
  const int col0 = blockIdx.x * 64;

  __shared__ __align__(16) float bt[2][16][64];

  // Staging assignment: thread t copies B[k0 + (t>>4)][col0 + (t&15)*4 .. +3]
  const int srow = threadIdx.x >> 4;        // 0..15
  const int scol = (threadIdx.x & 15) * 4;  // 0..60

  v8f acc0 = {}, acc1 = {}, acc2 = {}, acc3 = {};
  const float* __restrict__ Arow = A + (size_t)(row0 + n16) * K;

  const int nchunk = K / 16;

  // Prologue: stage chunk 0 into buffer 0.
  async_b128_to_lds(&bt[0][srow][scol],
                    B + (size_t)srow * N + col0 + scol);

  for (int c = 0; c < nchunk; ++c) {
    const int cur = c & 1;

    if (c + 1 < nchunk) {
      async_b128_to_lds(&bt[cur ^ 1][srow][scol],
                        B + (size_t)((c + 1) * 16 + srow) * N + col0 + scol);
      wait_async_le1();   // chunk c complete (async loads retire in order)
    } else {
      wait_async_le0();
    }
    __syncthreads();      // all waves' staging visible

    if (c + 4 < nchunk)   // prefetch A stream 4 chunks (256B) ahead
      __builtin_prefetch(Arow + (c + 4) * 16, 0, 1);

#pragma unroll
    for (int ks = 0; ks < 4; ++ks) {
      const int ka = c * 16 + ks * 4 + 2 * hi;   // global k for A fragment
      const int kk = ks * 4 + 2 * hi;            // k within LDS chunk
      v2f a; a.x = Arow[ka]; a.y = Arow[ka + 1];
      v2f b0; b0.x = bt[cur][kk][n16];      b0.y = bt[cur][kk + 1][n16];
      v2f b1; b1.x = bt[cur][kk][n16 + 16]; b1.y = bt[cur][kk + 1][n16 + 16];
      v2f b2; b2.x = bt[cur][kk][n16 + 32]; b2.y = bt[cur][kk + 1][n16 + 32];
      v2f b3; b3.x = bt[cur][kk][n16 + 48]; b3.y = bt[cur][kk + 1][n16 + 48];
      acc0 = wmma_f32(a, b0, acc0);
      acc1 = wmma_f32(a, b1, acc1);
      acc2 = wmma_f32(a, b2, acc2);
      acc3 = wmma_f32(a, b3, acc3);
    }
    __syncthreads();      // everyone done reading bt[cur] before re-stage
  }

#pragma unroll
  for (int j = 0; j < 8; ++j) {
    float* Cr = C + (size_t)(row0 + j + 8 * hi) * N + col0 + n16;
    Cr[0]  = acc0[j];
    Cr[16] = acc1[j];
    Cr[32] = acc2[j];
    Cr[48] = acc3[j];
  }
}

// ---------------------------------------------------------------------------
// RoPE, fp32, in place. X: (BATCH*SEQ, nheads*128) row-major token-major.
// One thread per (token, head, d<64) pair; rotates the (d, d+64) pair.
// angle = pos * theta^(-2d/128)
// ---------------------------------------------------------------------------
__global__ void rope_kernel(float* __restrict__ X, int nheads) {
  const int idx   = blockIdx.x * blockDim.x + threadIdx.x;
  const int total = BATCH * SEQ * nheads * 64;
  if (idx >= total) return;
  const int d = idx & 63;
  const int h = (idx >> 6) % nheads;
  const int t = idx / (64 * nheads);
  const int pos = t % SEQ;

  // theta^(-2d/128) = exp(-(2d/128)*ln(10000))
  const float inv = expf(-(float)(2 * d) * (1.0f / 128.0f) * 9.210340371976184f);
  const float ang = (float)pos * inv;
  float sv, cv;
  sincosf(ang, &sv, &cv);

  const size_t base = (size_t)t * (nheads * HD) + (size_t)h * HD + d;
  const float x1 = X[base];
  const float x2 = X[base + 64];
  X[base]      = x1 * cv - x2 * sv;
  X[base + 64] = x2 * cv + x1 * sv;
}

// ---------------------------------------------------------------------------
// Flash attention (causal, GQA 4:1). One wave per (b, h, 16-query tile).
// Q kept in registers as 32 A-fragments; streaming softmax with width-16
// shuffle reductions (rows of the D layout live in half-wave lane groups);
// P tile routed through a 16x17 LDS tile to re-shape D-layout -> A-layout.
// No divergent branches: EXEC is all-ones at every WMMA.
// ---------------------------------------------------------------------------
__global__ void flash_attn(const float* __restrict__ Q,
                           const float* __restrict__ K,
                           const float* __restrict__ V,
                           float* __restrict__ O) {
  const int lane = threadIdx.x & 31;
  const int n16  = lane & 15;
  const int hi   = lane >> 4;
  const int qt = blockIdx.x;
  const int h  = blockIdx.y;
  const int b  = blockIdx.z;
  const int kh = h / (NH / NKV);
  const int q0 = qt * 16;

  __shared__ float pbuf[16][17];

  // Load this wave's Q tile (16x128) as 32 A-fragments.
  const float* Qb = Q + (size_t)(b * SEQ + q0 + n16) * HID + (size_t)h * HD;
  v2f qf[32];
#pragma unroll
  for (int kc = 0; kc < 32; ++kc) {
    const int d = kc * 4 + 2 * hi;
    qf[kc].x = Qb[d];
    qf[kc].y = Qb[d + 1];
  }

  v8f o[8];
#pragma unroll
  for (int c = 0; c < 8; ++c) o[c] = (v8f){};
  float mrow[8], lrow[8];
#pragma unroll
  for (int j = 0; j < 8; ++j) { mrow[j] = -1.0e30f; lrow[j] = 0.0f; }

  const float scale = 0.08838834764831845f;  // 128^-0.5

  for (int k0 = 0; k0 <= q0; k0 += 16) {
    // ---- S = Q @ K^T  (B fragment: B[kk][n] = K[k0+n][kk]) ----
    const float* Kb = K + (size_t)(b * SEQ + k0 + n16) * KVD + (size_t)kh * HD;
    v8f s = {};
#pragma unroll
    for (int kc = 0; kc < 32; ++kc) {
      const int d = kc * 4 + 2 * hi;
      v2f bf; bf.x = Kb[d]; bf.y = Kb[d + 1];
      s = wmma_f32(qf[kc], bf, s);
    }

    // ---- mask + streaming softmax (row m = j + 8*hi, key col = n16) ----
#pragma unroll
    for (int j = 0; j < 8; ++j) {
      const int m = j + 8 * hi;
      float sv = s[j] * scale;
      sv = (k0 + n16 > q0 + m) ? -1.0e30f : sv;

      float tmax = sv;
      for (int off = 8; off >= 1; off >>= 1)
        tmax = fmaxf(tmax, __shfl_xor(tmax, off, 16));
      const float mnew = fmaxf(mrow[j], tmax);
      const float corr = expf(mrow[j] - mnew);   // first tile: exp(-1e30)=0
      const float p    = expf(sv - mnew);        // masked: underflows to 0

      float tsum = p;
      for (int off = 8; off >= 1; off >>= 1)
        tsum += __shfl_xor(tsum, off, 16);

      lrow[j] = lrow[j] * corr + tsum;
      mrow[j] = mnew;
#pragma unroll
      for (int c = 0; c < 8; ++c) o[c][j] *= corr;
      pbuf[m][n16] = p;
    }
    __syncthreads();

    // ---- O += P @ V  (A = P from LDS; B[kk][n] = V[k0+kk][dim n]) ----
    const float* Vb = V + (size_t)(b * SEQ + k0) * KVD + (size_t)kh * HD;
#pragma unroll
    for (int kc = 0; kc < 4; ++kc) {
      const int kk = kc * 4 + 2 * hi;
      v2f a; a.x = pbuf[n16][kk]; a.y = pbuf[n16][kk + 1];
#pragma unroll
      for (int c = 0; c < 8; ++c) {
        const float* V0 = Vb + (size_t)kk * KVD + c * 16 + n16;
        v2f bf; bf.x = V0[0]; bf.y = V0[KVD];
        o[c] = wmma_f32(a, bf, o[c]);
      }
    }
    __syncthreads();
  }

  // ---- normalize + store ----
  float* Ob = O + (size_t)(b * SEQ + q0) * HID + (size_t)h * HD;
#pragma unroll
  for (int j = 0; j < 8; ++j) {
    const int m = j + 8 * hi;
    const float inv = 1.0f / lrow[j];
#pragma unroll
    for (int c = 0; c < 8; ++c)
      Ob[(size_t)m * HID + c * 16 + n16] = o[c][j] * inv;
  }
}

// ---------------------------------------------------------------------------
// Launch. Inputs: x, mask(ignored: exactly causal), Wq, Wk, Wv, Wo.
// Workspace layout: Q (64MB) | K (16MB) | V (16MB) | attn (64MB) = 160MB.
// ---------------------------------------------------------------------------
extern "C" void kernel_launch(void* const* d_in, const int* in_sizes, int n_in,
                              void* d_out, int out_size, void* d_ws, size_t ws_size,
                              hipStream_t stream) {
  const float* x  = (const float*)d_in[0];
  const float* Wq = (const float*)d_in[2];
  const float* Wk = (const float*)d_in[3];
  const float* Wv = (const float*)d_in[4];
  const float* Wo = (const float*)d_in[5];
  float* out = (float*)d_out;

  const int Mtok = BATCH * SEQ;  // 4096 tokens

  float* Qb = (float*)d_ws;
  float* Kb = Qb + (size_t)Mtok * HID;
  float* Vb = Kb + (size_t)Mtok * KVD;
  float* Ab = Vb + (size_t)Mtok * KVD;

  const dim3 gblk(256, 1, 1);

  // Projections (block tile 128x64, 8 waves)
  gemm_f32_wmma<<<dim3(HID / 64, Mtok / 128), gblk, 0, stream>>>(x, Wq, Qb, Mtok, HID, HID);
  gemm_f32_wmma<<<dim3(KVD / 64, Mtok / 128), gblk, 0, stream>>>(x, Wk, Kb, Mtok, KVD, HID);
  gemm_f32_wmma<<<dim3(KVD / 64, Mtok / 128), gblk, 0, stream>>>(x, Wv, Vb, Mtok, KVD, HID);

  // RoPE on Q and K
  {
    const int nq = BATCH * SEQ * NH * 64;
    rope_kernel<<<(nq + 255) / 256, 256, 0, stream>>>(Qb, NH);
    const int nk = BATCH * SEQ * NKV * 64;
    rope_kernel<<<(nk + 255) / 256, 256, 0, stream>>>(Kb, NKV);
  }

  // Causal GQA flash attention
  flash_attn<<<dim3(SEQ / 16, NH, BATCH), dim3(32, 1, 1), 0, stream>>>(Qb, Kb, Vb, Ab);

  // Output projection
  gemm_f32_wmma<<<dim3(HID / 64, Mtok / 128), gblk, 0, stream>>>(Ab, Wo, out, Mtok, HID, HID);
}